// GCNConv_10788957847626
// MI455X (gfx1250) — compile-verified
//
#include <hip/hip_runtime.h>

typedef __attribute__((ext_vector_type(2))) float v2f;
typedef __attribute__((ext_vector_type(8))) float v8f;

#define D 64
#define LDS_PITCH 68   // 64 + 4 pad -> B reads hit distinct LDS banks

__global__ void init_deg_kernel(int* deg, int N) {
  int i = blockIdx.x * blockDim.x + threadIdx.x;
  if (i < N) deg[i] = 1;  // self loop counts as 1
}

__global__ void count_deg_kernel(const int* __restrict__ col, int* deg, int E) {
  int e = blockIdx.x * blockDim.x + threadIdx.x;
  if (e < E) atomicAdd(&deg[col[e]], 1);
}

__global__ void rsqrt_deg_kernel(const int* __restrict__ deg, float* dinv, int N) {
  int i = blockIdx.x * blockDim.x + threadIdx.x;
  if (i < N) dinv[i] = rsqrtf((float)deg[i]);  // deg >= 1, no clip needed
}

// h = x @ W^T in exact fp32 via V_WMMA_F32_16X16X4_F32.
// One wave per 16-row node tile; each wave produces the full 16x64 output tile
// as four 16x16 accumulators. W (64x64, 16KB) staged in LDS (padded pitch).
__global__ void __launch_bounds__(256) gemm_xwT_kernel(const float* __restrict__ x,
                                                       const float* __restrict__ W,
                                                       float* __restrict__ h,
                                                       int ntiles) {
  __shared__ float lw[D * LDS_PITCH];
  const int tid = threadIdx.x;
#pragma unroll
  for (int i = 0; i < 16; ++i) {           // 256 thr * 16 = 4096 = 64*64
    int idx = tid + i * 256;
    lw[(idx >> 6) * LDS_PITCH + (idx & 63)] = W[idx];
  }
  __syncthreads();

  const int wave = tid >> 5;
  const int lane = tid & 31;
  const int half = lane >> 4;              // 0: lanes 0-15, 1: lanes 16-31
  const int l16  = lane & 15;
  const int tile = blockIdx.x * 8 + wave;
  if (tile >= ntiles) return;              // wave-uniform: EXEC stays all-1s

  // A fragment (16x4 f32): lane<16 holds M=l16,K={k,k+1}; lane>=16 holds K={k+2,k+3}
  const float* xrow = x + ((size_t)(tile * 16 + l16)) * D + 2 * half;
  const int koff = 2 * half;

  v8f acc0 = {}, acc1 = {}, acc2 = {}, acc3 = {};
#pragma unroll
  for (int k = 0; k < D; k += 4) {
    v2f a  = *(const v2f*)(xrow + k);
    // B fragment (4x16 f32), B[k'][n] = W[n][k']: VGPR j -> K=j (lanes 0-15), K=j+2 (lanes 16-31)
    v2f b0 = *(const v2f*)(&lw[( 0 + l16) * LDS_PITCH + k + koff]);
    v2f b1 = *(const v2f*)(&lw[(16 + l16) * LDS_PITCH + k + koff]);
    v2f b2 = *(const v2f*)(&lw[(32 + l16) * LDS_PITCH + k + koff]);
    v2f b3 = *(const v2f*)(&lw[(48 + l16) * LDS_PITCH + k + koff]);
    acc0 = __builtin_amdgcn_wmma_f32_16x16x4_f32(false, a, false, b0, (short)0, acc0, false, false);
    acc1 = __builtin_amdgcn_wmma_f32_16x16x4_f32(false, a, false, b1, (short)0, acc1, false, false);
    acc2 = __builtin_amdgcn_wmma_f32_16x16x4_f32(false, a, false, b2, (short)0, acc2, false, false);
    acc3 = __builtin_amdgcn_wmma_f32_16x16x4_f32(false, a, false, b3, (short)0, acc3, false, false);
  }

  // C/D layout: VGPR i -> row M=i (lanes 0-15) / M=i+8 (lanes 16-31), N = l16
  float* hrow = h + (size_t)tile * 16 * D;
#pragma unroll
  for (int i = 0; i < 8; ++i) {
    int r = i + 8 * half;
    hrow[r * D +  0 + l16] = acc0[i];
    hrow[r * D + 16 + l16] = acc1[i];
    hrow[r * D + 32 + l16] = acc2[i];
    hrow[r * D + 48 + l16] = acc3[i];
  }
}

// out[n][f] = h[n][f] * dinv[n]^2  (self-loop message; initializes all of d_out)
__global__ void selfloop_kernel(const float* __restrict__ h, const float* __restrict__ dinv,
                                float* __restrict__ out, int total) {
  int i = blockIdx.x * blockDim.x + threadIdx.x;
  if (i < total) {
    float di = dinv[i >> 6];
    out[i] = h[i] * di * di;
  }
}

// 64 threads per edge: coalesced row gather of h, hardware f32 atomics into out (L2-resident).
__global__ void scatter_kernel(const int* __restrict__ ei, const float* __restrict__ ew,
                               const float* __restrict__ dinv, const float* __restrict__ h,
                               float* __restrict__ out, int E) {
  unsigned gid = blockIdx.x * 256u + threadIdx.x;
  unsigned e = gid >> 6;
  if (e >= (unsigned)E) return;
  int f = gid & 63;
  int r = ei[e];       // row
  int c = ei[E + e];   // col
  float norm = dinv[r] * ew[e] * dinv[c];
  unsafeAtomicAdd(&out[(size_t)c * D + f], h[(size_t)r * D + f] * norm);
}

extern "C" void kernel_launch(void* const* d_in, const int* in_sizes, int n_in,
                              void* d_out, int out_size, void* d_ws, size_t ws_size,
                              hipStream_t stream) {
  const float* x  = (const float*)d_in[0];
  const int*   ei = (const int*)d_in[1];   // [2, E], integer inputs arrive as int32
  const float* ew = (const float*)d_in[2];
  const float* W  = (const float*)d_in[3];
  float* out = (float*)d_out;

  const int N = in_sizes[0] / D;   // 100000
  const int E = in_sizes[2];       // 1600000

  char* ws = (char*)d_ws;
  size_t degBytes = (((size_t)N * sizeof(int)) + 255) & ~(size_t)255;
  int*   deg  = (int*)ws;
  float* dinv = (float*)(ws + degBytes);
  float* h    = (float*)(ws + 2 * degBytes);   // N*64 floats (25.6 MB)

  init_deg_kernel <<<(N + 255) / 256, 256, 0, stream>>>(deg, N);
  count_deg_kernel<<<(E + 255) / 256, 256, 0, stream>>>(ei + E, deg, E);
  rsqrt_deg_kernel<<<(N + 255) / 256, 256, 0, stream>>>(deg, dinv, N);

  int ntiles = (N + 15) / 16;                  // 6250 (exact)
  gemm_xwT_kernel<<<(ntiles + 7) / 8, 256, 0, stream>>>(x, W, h, ntiles);

  selfloop_kernel<<<(N * D + 255) / 256, 256, 0, stream>>>(h, dinv, out, N * D);

  unsigned total = (unsigned)E * 64u;          // 102.4M threads
  scatter_kernel<<<(total + 255u) / 256u, 256, 0, stream>>>(ei, ew, dinv, h, out, E);
}